// Gating_90735479095715
// MI455X (gfx1250) — compile-verified
//
#include <hip/hip_runtime.h>

typedef float v2f __attribute__((ext_vector_type(2)));
typedef float v8f __attribute__((ext_vector_type(8)));

#define B_ 4
#define S_ 4096
#define H_ 2048
#define E_ 8
#define TOPK_ 2
#define NTOK (B_ * S_)          // 16384 tokens
#define WAVES 8                 // waves per block (wave32 -> 256 threads)
#define TILES (NTOK / 16)       // 1024 16-token tiles

// One wave computes a 16(token) x 16(expert-col) f32 tile via
// V_WMMA_F32_16X16X4_F32 over K = H. Expert columns 8..15 are duplicates of
// 0..7 (W row index masked) so the K-loop needs no predication and EXEC stays
// all-ones as WMMA requires.
__global__ __launch_bounds__(256) void gating_topk_kernel(
    const float* __restrict__ x,      // [NTOK, H]
    const float* __restrict__ W,      // [E, H]
    const float* __restrict__ bias,   // [E]
    float* __restrict__ sparse_out,   // [NTOK, E]
    int*   __restrict__ idx_out,      // [NTOK, TOPK]
    float* __restrict__ gate_out)     // [NTOK, E]
{
    __shared__ float lds_logits[WAVES][16][E_];   // 4 KB: C-fragment transpose

    const int tid  = threadIdx.x;
    const int w    = tid >> 5;          // wave in block
    const int lane = tid & 31;
    const int g    = lane >> 4;         // K half-group (0/1)
    const int m    = lane & 15;         // A: token-in-tile; B/C: expert column

    const int  tile = blockIdx.x * WAVES + w;     // 0..TILES-1
    const long r0   = (long)tile * 16;            // first token of tile

    // Fold bias into the accumulator init (cols >= 8 hold duplicates, ignored).
    const float binit = bias[m & 7];
    v8f c;
    #pragma unroll
    for (int i = 0; i < 8; ++i) c[i] = binit;

    // Per-lane row pointers. A: x[r0+m][2g + ...]; B: W[m&7][2g + ...]
    const float* xrow = x + (r0 + m) * (long)H_ + 2 * g;
    const float* wrow = W + (long)(m & 7) * H_ + 2 * g;

    for (int kk = 0; kk < H_; kk += 64) {
        // gfx1250 global_prefetch_b8, ~4 KB ahead in the x stream
        __builtin_prefetch(xrow + kk + 1024, 0, 1);
        #pragma unroll
        for (int k4 = 0; k4 < 64; k4 += 4) {
            const int k = kk + k4;
            v2f a = *(const v2f*)(xrow + k);   // token m, K = k+2g+{0,1}
            v2f b = *(const v2f*)(wrow + k);   // expert m&7, same K pair
            // D = A(16x4) * B(4x16) + C  (f32, round-to-nearest-even)
            c = __builtin_amdgcn_wmma_f32_16x16x4_f32(
                    /*neg_a=*/false, a, /*neg_b=*/false, b,
                    /*c_mod=*/(short)0, c, /*reuse_a=*/false, /*reuse_b=*/false);
        }
    }

    // C layout: VGPR i, lanes 0-15 -> token i, expert=lane; lanes 16-31 ->
    // token 8+i, expert=lane-16. Stash real experts (cols < 8) in LDS so each
    // of lanes 0..15 can own one token's 8 logits.
    if (m < E_) {
        #pragma unroll
        for (int i = 0; i < 8; ++i)
            lds_logits[w][i + 8 * g][m] = c[i];
    }
    __syncthreads();

    if (lane < 16) {
        const long row = r0 + lane;
        float v[E_];
        #pragma unroll
        for (int e = 0; e < E_; ++e) v[e] = lds_logits[w][lane][e];

        // dense gate logits [B*S, E]
        float4 lo = {v[0], v[1], v[2], v[3]};
        float4 hi = {v[4], v[5], v[6], v[7]};
        *(float4*)(gate_out + row * E_)     = lo;
        *(float4*)(gate_out + row * E_ + 4) = hi;

        // top-2 with JAX tie-break (strict > keeps lowest index)
        int i0 = 0; float m0 = v[0];
        #pragma unroll
        for (int e = 1; e < E_; ++e) { if (v[e] > m0) { m0 = v[e]; i0 = e; } }
        int i1 = -1; float m1 = -__builtin_inff();
        #pragma unroll
        for (int e = 0; e < E_; ++e) { if (e != i0 && v[e] > m1) { m1 = v[e]; i1 = e; } }

        // sparse logits: -inf except the two winners (selects, no scratch)
        float s[E_];
        #pragma unroll
        for (int e = 0; e < E_; ++e)
            s[e] = (e == i0) ? m0 : ((e == i1) ? m1 : -__builtin_inff());
        float4 s0 = {s[0], s[1], s[2], s[3]};
        float4 s1 = {s[4], s[5], s[6], s[7]};
        *(float4*)(sparse_out + row * E_)     = s0;
        *(float4*)(sparse_out + row * E_ + 4) = s1;

        int2 ii = {i0, i1};
        *(int2*)(idx_out + row * TOPK_) = ii;
    }
}

extern "C" void kernel_launch(void* const* d_in, const int* in_sizes, int n_in,
                              void* d_out, int out_size, void* d_ws, size_t ws_size,
                              hipStream_t stream) {
    (void)in_sizes; (void)n_in; (void)out_size; (void)d_ws; (void)ws_size;
    const float* x    = (const float*)d_in[0];   // [B,S,H] f32
    const float* W    = (const float*)d_in[1];   // [E,H]  f32
    const float* bias = (const float*)d_in[2];   // [E]    f32

    // d_out = concat(sparse_logits f32 [B,S,E], indices i32 [B,S,2],
    //                gate_logit f32 [B*S,E]) flat in return order.
    float* sparse = (float*)d_out;
    int*   idx    = (int*)((float*)d_out + (size_t)NTOK * E_);
    float* gate   = (float*)d_out + (size_t)NTOK * E_ + (size_t)NTOK * TOPK_;

    dim3 grid(TILES / WAVES);   // 128 blocks x 8 waves = 1024 wave-tiles
    gating_topk_kernel<<<grid, 256, 0, stream>>>(x, W, bias, sparse, idx, gate);
}